// FGL_27376121544985
// MI455X (gfx1250) — compile-verified
//
#include <hip/hip_runtime.h>

// Problem constants (from reference)
#define N_B   8
#define INC   64
#define INN   20000
#define OUTC  64
#define OUTN  8000
#define MAXD  32
#define OTILE 32   // outputs per block

typedef __attribute__((ext_vector_type(2))) float v2f;
typedef __attribute__((ext_vector_type(8))) float v8f;

// ---------------------------------------------------------------------------
// Kernel 0: W[k,c] = ct_g[k] * ct_v[k,c] / ||ct_v[k,:]||   (64x64, 16 KB)
// ---------------------------------------------------------------------------
__global__ void wnorm_kernel(const float* __restrict__ ct_v,
                             const float* __restrict__ ct_g,
                             float* __restrict__ W) {
    const int k = threadIdx.x;          // 64 threads, one row each
    float s = 0.f;
    #pragma unroll 8
    for (int c = 0; c < INC; ++c) {
        float v = ct_v[k * INC + c];
        s = __builtin_fmaf(v, v, s);
    }
    const float scale = ct_g[k] * rsqrtf(s);
    #pragma unroll 8
    for (int c = 0; c < INC; ++c)
        W[k * INC + c] = ct_v[k * INC + c] * scale;
}

// ---------------------------------------------------------------------------
// Fused gather-pool + 64x64 WMMA projection + bias.
// grid = (OUTN/OTILE, N_B), block = 256 threads (8 x wave32)
// ---------------------------------------------------------------------------
__global__ __launch_bounds__(256)
void fgl_fused_kernel(const float* __restrict__ x,      // (N,INC,INN)
                      const float* __restrict__ wgt,    // (INC,INN)
                      const float* __restrict__ Wg,     // (OUTC,INC) normalized
                      const float* __restrict__ bias,   // (1,OUTC,OUTN)
                      const float* __restrict__ mask,   // (OUTN,MAXD)
                      const int*   __restrict__ Aidx,   // (OUTN,MAXD)
                      float*       __restrict__ out) {  // (N,OUTC,OUTN)
    __shared__ int   sA[OTILE][MAXD + 1];   // padded: kill stride-32 conflicts
    __shared__ float sM[OTILE][MAXD + 1];
    __shared__ float sW[OUTC][INC + 1];     // padded: kill stride-64 conflicts
    __shared__ float sP[INC][OTILE + 1];    // pooled tile [c][o], padded

    const int tid   = threadIdx.x;
    const int b     = blockIdx.y;
    const int obase = blockIdx.x * OTILE;

    // ---- stage in A / mask tile and the 64x64 projection matrix ----
    for (int i = tid; i < OTILE * MAXD; i += 256) {
        const int o = i >> 5, d = i & (MAXD - 1);
        sA[o][d] = Aidx[(obase + o) * MAXD + d];
        sM[o][d] = mask[(obase + o) * MAXD + d];
    }
    for (int i = tid; i < OUTC * INC; i += 256)
        sW[i >> 6][i & 63] = Wg[i];
    __syncthreads();

    // ---- phase 1: masked gather-reduce into pooled[c][o] ----
    // Each thread owns ONE output column o = tid&31 and 8 channels
    // c = (tid>>5) + 8p.  d is the outer loop so each index/mask is read
    // from LDS exactly once; each d issues 16 independent global gathers
    // whose c-strides fold into 24-bit instruction offsets (p*640000 B).
    {
        const int o     = tid & (OTILE - 1);
        const int cbase = tid >> 5;               // 0..7
        const float* __restrict__ xp = x   + ((size_t)b * INC + cbase) * INN;
        const float* __restrict__ wp = wgt + (size_t)cbase * INN;

        float acc[8] = {0.f, 0.f, 0.f, 0.f, 0.f, 0.f, 0.f, 0.f};
        #pragma unroll 2
        for (int d = 0; d < MAXD; ++d) {
            const int   idx = sA[o][d];
            const float m   = sM[o][d];
            #pragma unroll
            for (int p = 0; p < 8; ++p) {
                const float wv = wp[(size_t)p * 8 * INN + idx];
                const float xv = xp[(size_t)p * 8 * INN + idx];
                acc[p] = __builtin_fmaf(m * wv, xv, acc[p]);
            }
        }
        #pragma unroll
        for (int p = 0; p < 8; ++p)
            sP[cbase + 8 * p][o] = acc[p];
    }
    __syncthreads();

    // ---- phase 2: y[k0:k0+16, o0:o0+16] = W(16x64) * pooled(64x16) via WMMA ----
    // 8 waves: wave = kt (0..3) x ot (0..1); 16 x V_WMMA_F32_16X16X4_F32 each.
    const int lane = tid & 31;
    const int wave = tid >> 5;
    const int k0   = (wave & 3) << 4;       // k-tile base
    const int o0   = (wave >> 2) << 4;      // o-subtile base within OTILE
    const int lhi  = lane >> 4;             // half-wave select
    const int llo  = lane & 15;

    v8f acc = {};
    #pragma unroll
    for (int c = 0; c < INC; c += 4) {
        // A fragment (16x4 f32): lanes 0-15 -> K={c,c+1}, lanes 16-31 -> K={c+2,c+3}
        v2f a;
        a.x = sW[k0 + llo][c + 2 * lhi + 0];
        a.y = sW[k0 + llo][c + 2 * lhi + 1];
        // B fragment (4x16 f32): mirrored layout, N striped across lanes
        v2f bf;
        bf.x = sP[c + 2 * lhi + 0][o0 + llo];
        bf.y = sP[c + 2 * lhi + 1][o0 + llo];
        acc = __builtin_amdgcn_wmma_f32_16x16x4_f32(
            /*neg_a=*/false, a, /*neg_b=*/false, bf,
            /*c_mod=*/(short)0, acc, /*reuse_a=*/false, /*reuse_b=*/false);
    }

    // ---- epilogue: C layout (VGPR r, lane l): M = r + 8*lhi, N = llo ----
    #pragma unroll
    for (int r = 0; r < 8; ++r) {
        const int k = k0 + r + 8 * lhi;
        const int o = obase + o0 + llo;
        out[((size_t)b * OUTC + k) * OUTN + o] = acc[r] + bias[(size_t)k * OUTN + o];
    }
}

// ---------------------------------------------------------------------------
extern "C" void kernel_launch(void* const* d_in, const int* in_sizes, int n_in,
                              void* d_out, int out_size, void* d_ws, size_t ws_size,
                              hipStream_t stream) {
    const float* x    = (const float*)d_in[0];
    const float* wgt  = (const float*)d_in[1];
    const float* ct_v = (const float*)d_in[2];
    const float* ct_g = (const float*)d_in[3];
    const float* bias = (const float*)d_in[4];
    const float* mask = (const float*)d_in[5];
    const int*   Aidx = (const int*)d_in[6];
    float* out = (float*)d_out;
    float* Wg  = (float*)d_ws;           // 64*64*4 = 16 KB scratch

    wnorm_kernel<<<1, 64, 0, stream>>>(ct_v, ct_g, Wg);

    dim3 grid(OUTN / OTILE, N_B);        // (250, 8)
    fgl_fused_kernel<<<grid, 256, 0, stream>>>(x, wgt, Wg, bias, mask, Aidx, out);
}